// MANO1D_69612829934184
// MI455X (gfx1250) — compile-verified
//
#include <hip/hip_runtime.h>

// ---------------------------------------------------------------------------
// Galerkin linear attention, algebraically collapsed for MI455X (gfx1250).
//   S[b]    = keys[b]^T @ values[b]                       (64x64, reduce N)
//   Weff[b] = (1/N) * sum_h Wq_h^T (Wk_h S[b] Wk_h^T) Wout_h^T   (64x64)
//   out     = queries[b] @ Weff[b] + bout
// All matmuls on v_wmma_f32_16x16x4_f32 (exact fp32). Bandwidth-bound:
// ~67MB traffic -> ~2.9us floor at 23.3 TB/s; compute ~1.1 GFLOP.
// ---------------------------------------------------------------------------

#define B_      4
#define N_      16384
#define DIN     64
#define H_      8
#define INNER   512
#define PAD     72                 // padded LDS row stride (floats)
#define CHUNKS  32                 // partial-S chunks per batch
#define RPW     (N_ / CHUNKS)      // rows per workgroup = 512
#define TR      64                 // staged tile rows
#define NT      (RPW / TR)         // tiles per workgroup = 8

typedef __attribute__((ext_vector_type(2))) float v2f;
typedef __attribute__((ext_vector_type(8))) float v8f;

__device__ __forceinline__ v8f wmma4(v2f a, v2f b, v8f c) {
  // D = A(16x4,f32) x B(4x16,f32) + C(16x16,f32)
  return __builtin_amdgcn_wmma_f32_16x16x4_f32(false, a, false, b, (short)0, c,
                                               false, false);
}

// Generic LDS pointer -> 32-bit LDS byte offset (flat LDS aperture keeps the
// workgroup-relative offset in the low 32 bits).
__device__ __forceinline__ unsigned lds_off(const void* p) {
  return (unsigned)(uintptr_t)p;
}

__device__ __forceinline__ void async_b128(unsigned lds_addr, const float* g) {
  asm volatile("global_load_async_to_lds_b128 %0, %1, off"
               :: "v"(lds_addr), "v"(g) : "memory");
}
__device__ __forceinline__ void wait_async0() {
  asm volatile("s_wait_asynccnt 0" ::: "memory");
}

// ---------------------------------------------------------------------------
// Kernel 1: partial S[b] over a 512-row chunk. Double-buffered async staging.
// grid (CHUNKS, B), 256 threads (8 waves). Wave w owns output tiles
// (tm = w>>1, tn = (w&1)*2 + {0,1}) of the 64x64 S.
// ---------------------------------------------------------------------------
__global__ void __launch_bounds__(256)
k_partial_S(const float* __restrict__ keys, const float* __restrict__ values,
            float* __restrict__ part) {
  __shared__ float Lk[2][TR * PAD];
  __shared__ float Lv[2][TR * PAD];

  const int b = blockIdx.y, chunk = blockIdx.x;
  const int tid = threadIdx.x;
  const int wave = tid >> 5, lane = tid & 31;
  const int lrow = lane >> 4, lcol = lane & 15;
  const int tm = wave >> 1, tn0 = (wave & 1) * 2;

  const long base = ((long)b * N_ + (long)chunk * RPW) * DIN;
  const int srow = tid >> 2;          // staging row 0..63
  const int scol = (tid & 3) * 16;    // staging col 0/16/32/48

#define STAGE_TILE(tile, buf)                                                \
  {                                                                          \
    const float* gk = keys + base + ((long)(tile) * TR + srow) * DIN + scol; \
    const float* gv = values + base + ((long)(tile) * TR + srow) * DIN + scol;\
    float* lk = &Lk[(buf)][srow * PAD + scol];                               \
    float* lv = &Lv[(buf)][srow * PAD + scol];                               \
    _Pragma("unroll") for (int q = 0; q < 4; ++q) {                          \
      async_b128(lds_off(lk + q * 4), gk + q * 4);                           \
      async_b128(lds_off(lv + q * 4), gv + q * 4);                           \
    }                                                                        \
  }

  v8f acc0 = {}, acc1 = {};
  STAGE_TILE(0, 0)

  for (int t = 0; t < NT; ++t) {
    wait_async0();        // own tile-t async copies landed in LDS
    __syncthreads();      // everyone's tile-t landed; tile-(t-1) reads done
    if (t + 1 < NT) STAGE_TILE(t + 1, (t + 1) & 1)   // overlap next loads

    const float* lk = &Lk[t & 1][0];
    const float* lv = &Lv[t & 1][0];
#pragma unroll
    for (int ks = 0; ks < TR / 4; ++ks) {
      const int kk = ks * 4 + lrow * 2;
      v2f a, b0, b1;
      a.x  = lk[kk * PAD + tm * 16 + lcol];
      a.y  = lk[(kk + 1) * PAD + tm * 16 + lcol];
      b0.x = lv[kk * PAD + tn0 * 16 + lcol];
      b0.y = lv[(kk + 1) * PAD + tn0 * 16 + lcol];
      b1.x = lv[kk * PAD + (tn0 + 1) * 16 + lcol];
      b1.y = lv[(kk + 1) * PAD + (tn0 + 1) * 16 + lcol];
      acc0 = wmma4(a, b0, acc0);
      acc1 = wmma4(a, b1, acc1);
    }
  }
#undef STAGE_TILE

  float* po = part + (long)(b * CHUNKS + chunk) * 4096;
#pragma unroll
  for (int v = 0; v < 8; ++v) {
    const int row = tm * 16 + v + 8 * lrow;          // C/D layout: M = v + 8*(lane>>4)
    po[row * 64 + tn0 * 16 + lcol]       = acc0[v];
    po[row * 64 + (tn0 + 1) * 16 + lcol] = acc1[v];
  }
}

// ---------------------------------------------------------------------------
// Kernel 2: deterministic fixed-order reduction of CHUNKS partials -> S.
// ---------------------------------------------------------------------------
__global__ void __launch_bounds__(256)
k_reduce_S(const float* __restrict__ part, float* __restrict__ S) {
  const int gid = blockIdx.x * 256 + threadIdx.x;    // 0..16383
  const int b = gid >> 12, e = gid & 4095;
  float s = 0.f;
  for (int c = 0; c < CHUNKS; ++c)
    s += part[(long)(b * CHUNKS + c) * 4096 + e];
  S[gid] = s;
}

// ---------------------------------------------------------------------------
// Kernel 3: Weff[b] = (1/N) * sum_h Wq_h^T (Wk_h S Wk_h^T) Wout_h^T.
// One workgroup per batch; four chained 64x64 WMMA GEMMs through LDS per head.
// ---------------------------------------------------------------------------
__global__ void __launch_bounds__(256)
k_weff(const float* __restrict__ S, const float* __restrict__ Wq,
       const float* __restrict__ Wk, const float* __restrict__ Wout,
       float* __restrict__ Weff) {
  __shared__ float Sl[64 * PAD], Wkl[64 * PAD], Wql[64 * PAD], Wol[64 * PAD];
  __shared__ float M1[64 * PAD], M2[64 * PAD], M3[64 * PAD];

  const int b = blockIdx.x;
  const int tid = threadIdx.x;
  const int wave = tid >> 5, lane = tid & 31;
  const int lrow = lane >> 4, lcol = lane & 15;
  const int tm = wave >> 1, tn0 = (wave & 1) * 2;
  const int srow = tid >> 2, scol = (tid & 3) * 16;

#pragma unroll
  for (int q = 0; q < 16; ++q)
    Sl[srow * PAD + scol + q] = S[b * 4096 + srow * 64 + scol + q];

  v8f w0 = {}, w1 = {};   // persistent Weff accumulators across heads

  for (int h = 0; h < H_; ++h) {
    __syncthreads();      // previous head's reads of Wkl/Wql/Wol/M* done
#pragma unroll
    for (int q = 0; q < 16; ++q) {
      Wkl[srow * PAD + scol + q] = Wk[(h * 64 + srow) * DIN + scol + q];
      Wql[srow * PAD + scol + q] = Wq[(h * 64 + srow) * DIN + scol + q];
      Wol[srow * PAD + scol + q] = Wout[srow * INNER + h * 64 + scol + q];
    }
    __syncthreads();

    // GEMM1: M1 = Wk_h @ S
    {
      v8f c0 = {}, c1 = {};
#pragma unroll
      for (int ks = 0; ks < 16; ++ks) {
        const int kk = ks * 4 + lrow * 2;
        v2f a, b0, b1;
        a.x  = Wkl[(tm * 16 + lcol) * PAD + kk];
        a.y  = Wkl[(tm * 16 + lcol) * PAD + kk + 1];
        b0.x = Sl[kk * PAD + tn0 * 16 + lcol];
        b0.y = Sl[(kk + 1) * PAD + tn0 * 16 + lcol];
        b1.x = Sl[kk * PAD + (tn0 + 1) * 16 + lcol];
        b1.y = Sl[(kk + 1) * PAD + (tn0 + 1) * 16 + lcol];
        c0 = wmma4(a, b0, c0);
        c1 = wmma4(a, b1, c1);
      }
#pragma unroll
      for (int v = 0; v < 8; ++v) {
        const int row = tm * 16 + v + 8 * lrow;
        M1[row * PAD + tn0 * 16 + lcol]       = c0[v];
        M1[row * PAD + (tn0 + 1) * 16 + lcol] = c1[v];
      }
    }
    __syncthreads();

    // GEMM2: M2 = M1 @ Wk_h^T   (B(k,n) = Wk_h[n][k] -> row reads of Wkl)
    {
      v8f c0 = {}, c1 = {};
#pragma unroll
      for (int ks = 0; ks < 16; ++ks) {
        const int kk = ks * 4 + lrow * 2;
        v2f a, b0, b1;
        a.x  = M1[(tm * 16 + lcol) * PAD + kk];
        a.y  = M1[(tm * 16 + lcol) * PAD + kk + 1];
        b0.x = Wkl[(tn0 * 16 + lcol) * PAD + kk];
        b0.y = Wkl[(tn0 * 16 + lcol) * PAD + kk + 1];
        b1.x = Wkl[((tn0 + 1) * 16 + lcol) * PAD + kk];
        b1.y = Wkl[((tn0 + 1) * 16 + lcol) * PAD + kk + 1];
        c0 = wmma4(a, b0, c0);
        c1 = wmma4(a, b1, c1);
      }
#pragma unroll
      for (int v = 0; v < 8; ++v) {
        const int row = tm * 16 + v + 8 * lrow;
        M2[row * PAD + tn0 * 16 + lcol]       = c0[v];
        M2[row * PAD + (tn0 + 1) * 16 + lcol] = c1[v];
      }
    }
    __syncthreads();

    // GEMM3: M3 = M2 @ Wout_h^T  (B(e,o) = Wout_h[o][e])
    {
      v8f c0 = {}, c1 = {};
#pragma unroll
      for (int ks = 0; ks < 16; ++ks) {
        const int kk = ks * 4 + lrow * 2;
        v2f a, b0, b1;
        a.x  = M2[(tm * 16 + lcol) * PAD + kk];
        a.y  = M2[(tm * 16 + lcol) * PAD + kk + 1];
        b0.x = Wol[(tn0 * 16 + lcol) * PAD + kk];
        b0.y = Wol[(tn0 * 16 + lcol) * PAD + kk + 1];
        b1.x = Wol[((tn0 + 1) * 16 + lcol) * PAD + kk];
        b1.y = Wol[((tn0 + 1) * 16 + lcol) * PAD + kk + 1];
        c0 = wmma4(a, b0, c0);
        c1 = wmma4(a, b1, c1);
      }
#pragma unroll
      for (int v = 0; v < 8; ++v) {
        const int row = tm * 16 + v + 8 * lrow;
        M3[row * PAD + tn0 * 16 + lcol]       = c0[v];
        M3[row * PAD + (tn0 + 1) * 16 + lcol] = c1[v];
      }
    }
    __syncthreads();

    // GEMM4: Weff += Wq_h^T @ M3   (A(i,d) = Wq_h[d][i])
#pragma unroll
    for (int ks = 0; ks < 16; ++ks) {
      const int kk = ks * 4 + lrow * 2;
      v2f a, b0, b1;
      a.x  = Wql[kk * PAD + tm * 16 + lcol];
      a.y  = Wql[(kk + 1) * PAD + tm * 16 + lcol];
      b0.x = M3[kk * PAD + tn0 * 16 + lcol];
      b0.y = M3[(kk + 1) * PAD + tn0 * 16 + lcol];
      b1.x = M3[kk * PAD + (tn0 + 1) * 16 + lcol];
      b1.y = M3[(kk + 1) * PAD + (tn0 + 1) * 16 + lcol];
      w0 = wmma4(a, b0, w0);
      w1 = wmma4(a, b1, w1);
    }
  }

  const float inv = 1.0f / (float)N_;
#pragma unroll
  for (int v = 0; v < 8; ++v) {
    const int row = tm * 16 + v + 8 * lrow;
    Weff[b * 4096 + row * 64 + tn0 * 16 + lcol]       = w0[v] * inv;
    Weff[b * 4096 + row * 64 + (tn0 + 1) * 16 + lcol] = w1[v] * inv;
  }
}

// ---------------------------------------------------------------------------
// Kernel 4: out[b, r, :] = queries[b, r, :] @ Weff[b] + bout. 128 rows / WG.
// ---------------------------------------------------------------------------
__global__ void __launch_bounds__(256)
k_out(const float* __restrict__ queries, const float* __restrict__ Weff,
      const float* __restrict__ bout, float* __restrict__ out) {
  __shared__ float Ql[128 * PAD];
  __shared__ float Wl[64 * PAD];

  const int b = blockIdx.y, rt = blockIdx.x;
  const int tid = threadIdx.x;
  const int wave = tid >> 5, lane = tid & 31;
  const int lrow = lane >> 4, lcol = lane & 15;
  const int tm = wave;                         // 8 waves x 16 rows = 128 rows
  const int srow = tid >> 2, scol = (tid & 3) * 16;

  const long obase = (long)b * N_ + (long)rt * 128;
  const long qbase = obase * DIN;

  // async stage Weff[b] (64x64) and the 128x64 query tile
#pragma unroll
  for (int q = 0; q < 4; ++q)
    async_b128(lds_off(&Wl[srow * PAD + scol + q * 4]),
               Weff + b * 4096 + srow * 64 + scol + q * 4);
#pragma unroll
  for (int half = 0; half < 2; ++half) {
    const int row = half * 64 + srow;
#pragma unroll
    for (int q = 0; q < 4; ++q)
      async_b128(lds_off(&Ql[row * PAD + scol + q * 4]),
                 queries + qbase + (long)row * DIN + scol + q * 4);
  }
  wait_async0();
  __syncthreads();

  v8f acc[4] = {{}, {}, {}, {}};
#pragma unroll
  for (int ks = 0; ks < 16; ++ks) {
    const int kk = ks * 4 + lrow * 2;
    v2f a;
    a.x = Ql[(tm * 16 + lcol) * PAD + kk];
    a.y = Ql[(tm * 16 + lcol) * PAD + kk + 1];
#pragma unroll
    for (int tn = 0; tn < 4; ++tn) {
      v2f bb;
      bb.x = Wl[kk * PAD + tn * 16 + lcol];
      bb.y = Wl[(kk + 1) * PAD + tn * 16 + lcol];
      acc[tn] = wmma4(a, bb, acc[tn]);
    }
  }

#pragma unroll
  for (int tn = 0; tn < 4; ++tn) {
    const float bo = bout[tn * 16 + lcol];
#pragma unroll
    for (int v = 0; v < 8; ++v) {
      const int row = tm * 16 + v + 8 * lrow;
      out[(obase + row) * DIN + tn * 16 + lcol] = acc[tn][v] + bo;
    }
  }
}

// ---------------------------------------------------------------------------
extern "C" void kernel_launch(void* const* d_in, const int* in_sizes, int n_in,
                              void* d_out, int out_size, void* d_ws,
                              size_t ws_size, hipStream_t stream) {
  (void)in_sizes; (void)n_in; (void)out_size; (void)ws_size;
  const float* queries = (const float*)d_in[0];
  const float* keys    = (const float*)d_in[1];
  const float* values  = (const float*)d_in[2];
  const float* Wq      = (const float*)d_in[3];
  const float* Wk      = (const float*)d_in[4];
  const float* Wout    = (const float*)d_in[5];
  const float* bout    = (const float*)d_in[6];
  float* out = (float*)d_out;

  float* wsf  = (float*)d_ws;
  float* part = wsf;                                 // B*CHUNKS*4096 = 524288 f
  float* S    = wsf + (long)B_ * CHUNKS * 4096;      // 16384 f
  float* Weff = S + (long)B_ * 4096;                 // 16384 f  (~2.2 MB total)

  k_partial_S<<<dim3(CHUNKS, B_), 256, 0, stream>>>(keys, values, part);
  k_reduce_S<<<64, 256, 0, stream>>>(part, S);
  k_weff<<<B_, 256, 0, stream>>>(S, Wq, Wk, Wout, Weff);
  k_out<<<dim3(N_ / 128, B_), 256, 0, stream>>>(queries, Weff, bout, out);
}